// ConcatVolume_15487652069509
// MI455X (gfx1250) — compile-verified
//
#include <hip/hip_runtime.h>
#include <stdint.h>

// ConcatVolume for stereo cost volumes.
// fl, fr: (1, 32, 128, 256) fp32.  bins: (48,) int32.
// out: [vol_lr | vol_rl], each (1, 64, 48, 128, 256) fp32, concatenated flat.
//
// Store-bandwidth-bound (805 MB writes vs 8.4 MB reads). Strategy:
//  * one block per (c, h); stage the 1KB fl row + 1KB fr row into LDS with
//    CDNA5 async global->LDS copies (ASYNCcnt), reuse ~96x,
//  * four 64-thread groups emit one float4 non-temporal store per disparity,
//    fully coalesced along W.

typedef float v4f __attribute__((ext_vector_type(4)));
typedef __attribute__((address_space(3))) float lds_float;

#define CV_F 32
#define CV_H 128
#define CV_W 256

__device__ __forceinline__ void async_copy16(float* lds_dst, const float* gsrc) {
  // VDST = LDS byte address (32-bit), VADDR = 64-bit global address, no SADDR.
  uint32_t loff = (uint32_t)(uintptr_t)(lds_float*)lds_dst;
  asm volatile("global_load_async_to_lds_b128 %0, %1, off"
               :
               : "v"(loff), "v"(gsrc)
               : "memory");
}

__global__ __launch_bounds__(256) void concat_volume_kernel(
    const float* __restrict__ fl, const float* __restrict__ fr,
    const int* __restrict__ bins, float* __restrict__ out, int D)
{
  __shared__ __align__(16) float s_fl[CV_W];
  __shared__ __align__(16) float s_fr[CV_W];
  __shared__ int s_b[64];

  const int c = blockIdx.x & (CV_F - 1);  // channel 0..31
  const int h = blockIdx.x >> 5;          // row 0..127
  const int t = threadIdx.x;

  const float* flRow = fl + ((size_t)c * CV_H + h) * CV_W;
  const float* frRow = fr + ((size_t)c * CV_H + h) * CV_W;

  // Wave 0 (lanes 0..31, EXEC all ones) stages both rows: 4 x 512B async copies.
  if (t < 32) {
    const int o = t * 4;
    async_copy16(&s_fl[o],       flRow + o);
    async_copy16(&s_fl[o + 128], flRow + o + 128);
    async_copy16(&s_fr[o],       frRow + o);
    async_copy16(&s_fr[o + 128], frRow + o + 128);
    asm volatile("s_wait_asynccnt 0" ::: "memory");
  }
  if (t < D && t < 64) s_b[t] = bins[t];
  __syncthreads();

  const int r  = t >> 6;         // 0..3, uniform within each wave32
  const int w0 = (t & 63) << 2;  // float4 base column
  const size_t HW  = (size_t)CV_H * CV_W;          // d-stride in the volume
  const size_t VOL = (size_t)(2 * CV_F) * (size_t)D * HW;  // one volume
  const int chan = c + ((r & 1) << 5);             // c or c+32
  const size_t volSel = (size_t)(r >> 1);          // vol_lr or vol_rl

  float* p = out + volSel * VOL
           + ((size_t)chan * (size_t)D * CV_H + (size_t)h) * CV_W + w0;

  if (r == 0) {
    // vol_lr top: (w >= b) ? fl[w] : 0   (aligned, hoist the LDS read)
    const v4f base = *(const v4f*)&s_fl[w0];
    for (int d = 0; d < D; ++d) {
      const int b = s_b[d];
      v4f v;
#pragma unroll
      for (int j = 0; j < 4; ++j) v[j] = (w0 + j >= b) ? base[j] : 0.0f;
      __builtin_nontemporal_store(v, (v4f*)p);
      p += HW;
    }
  } else if (r == 1) {
    // vol_lr bottom: (w >= b) ? fr[w - b] : 0
    for (int d = 0; d < D; ++d) {
      const int b = s_b[d];
      v4f v;
#pragma unroll
      for (int j = 0; j < 4; ++j) {
        const int w = w0 + j;
        int idx = w - b;
        idx = idx < 0 ? 0 : idx;
        v[j] = (w >= b) ? s_fr[idx] : 0.0f;
      }
      __builtin_nontemporal_store(v, (v4f*)p);
      p += HW;
    }
  } else if (r == 2) {
    // vol_rl top: (w < W - b) ? fl[w + b] : 0
    for (int d = 0; d < D; ++d) {
      const int b = s_b[d];
      v4f v;
#pragma unroll
      for (int j = 0; j < 4; ++j) {
        const int w = w0 + j;
        int idx = w + b;
        idx = idx > (CV_W - 1) ? (CV_W - 1) : idx;
        v[j] = (w < CV_W - b) ? s_fl[idx] : 0.0f;
      }
      __builtin_nontemporal_store(v, (v4f*)p);
      p += HW;
    }
  } else {
    // vol_rl bottom: (w < W - b) ? fr[w] : 0   (aligned, hoist)
    const v4f base = *(const v4f*)&s_fr[w0];
    for (int d = 0; d < D; ++d) {
      const int b = s_b[d];
      v4f v;
#pragma unroll
      for (int j = 0; j < 4; ++j) v[j] = (w0 + j < CV_W - b) ? base[j] : 0.0f;
      __builtin_nontemporal_store(v, (v4f*)p);
      p += HW;
    }
  }
}

extern "C" void kernel_launch(void* const* d_in, const int* in_sizes, int n_in,
                              void* d_out, int out_size, void* d_ws, size_t ws_size,
                              hipStream_t stream) {
  const float* fl   = (const float*)d_in[0];
  const float* fr   = (const float*)d_in[1];
  const int*   bins = (const int*)d_in[2];
  const int D = in_sizes[2];  // 48

  dim3 grid(CV_F * CV_H);  // 4096 blocks: one per (channel, image row)
  dim3 block(256);
  concat_volume_kernel<<<grid, block, 0, stream>>>(fl, fr, bins, (float*)d_out, D);
}